// AnyQNN_19842748907786
// MI455X (gfx1250) — compile-verified
//
#include <hip/hip_runtime.h>
#include <stdint.h>

// Per-row vector quantization (nearest-of-16 codebook gather).
// x: [R=4096, C=2048] f32, values: [R, V=16] f32  ->  out: [R, C] f32
//
// Memory-bound (67 MB @ 23.3 TB/s ~ 2.9 us floor). Data path: two TDM
// tensor_load_to_lds DMAs per row (4 KB halves) pipelined against compute
// via split TENSORcnt waits (<=1, then 0). Waves consume LDS with
// ds_load_b128 and emit global_store_b128. Compute: argmin (x-v)^2 ==
// argmin (v^2 - 2xv): per candidate 1 v_fma vs precomputed (a=-2v, b=v^2)
// + v_cmp + 2 v_cndmask; strict '<' preserves argmin's first-min tie rule.

typedef __attribute__((ext_vector_type(4))) unsigned int v4u;
typedef __attribute__((ext_vector_type(8))) int          v8i;
typedef __attribute__((ext_vector_type(4))) int          v4i;

#define VQ_COLS  2048
#define VQ_HALF  1024
#define VQ_NV    16
#define VQ_BLOCK 256

// Issue one TDM 1-D tile load: nelem f32 from gsrc -> LDS byte offset lds_off.
__device__ __forceinline__ void tdm_load_f32(const float* gsrc, uint32_t lds_off,
                                             int nelem) {
  const uint64_t ga = (uint64_t)(uintptr_t)gsrc;

  // D# group 0: count=1, lds_addr, global_addr[56:0], type=2 ("image")
  v4u g0;
  g0.x = 1u;
  g0.y = lds_off;
  g0.z = (uint32_t)ga;
  g0.w = (uint32_t)((ga >> 32) & 0x01FFFFFFu) | (2u << 30);

  // D# group 1: data_size=4B; tensor_dim0=tile_dim0=nelem; dim1=1; stride0=nelem.
  v8i g1;
  g1[0] = (int)(2u << 16);                 // [17:16] data_size = 2 -> 4B
  g1[1] = (int)(((uint32_t)nelem) << 16);  // tensor_dim0[15:0] @ bits 63:48
  g1[2] = (int)(1u << 16);                 // tensor_dim1 = 1   @ bits 95:80
  g1[3] = (int)(((uint32_t)nelem) << 16);  // tile_dim0         @ bits 127:112
  g1[4] = 1;                               // tile_dim1 = 1     @ bits 143:128
  g1[5] = nelem;                           // tensor_dim0_stride[31:0]
  g1[6] = 0;
  g1[7] = 0;

  v4i g2 = {0, 0, 0, 0};                   // <=2D: groups 2/3 unused
  v4i g3 = {0, 0, 0, 0};
  v8i g4 = {0, 0, 0, 0, 0, 0, 0, 0};       // extra group (6-arg toolchain form)

  __builtin_amdgcn_tensor_load_to_lds(g0, g1, g2, g3, g4, 0);
}

__device__ __forceinline__ float nearest16(float x, const float* vv,
                                           const float* aa, const float* bb) {
  float bs = __builtin_fmaf(x, aa[0], bb[0]);
  float bv = vv[0];
#pragma unroll
  for (int i = 1; i < VQ_NV; ++i) {
    float s = __builtin_fmaf(x, aa[i], bb[i]);
    bool c = s < bs;
    bs = c ? s : bs;
    bv = c ? vv[i] : bv;
  }
  return bv;
}

__global__ __launch_bounds__(VQ_BLOCK) void vq_rows_kernel(
    const float* __restrict__ x, const float* __restrict__ values,
    float* __restrict__ out) {
  __shared__ float s_x[VQ_COLS];
  __shared__ float s_v[VQ_NV];

  const int row = blockIdx.x;
  const int tid = threadIdx.x;

  // Wave 0 queues both TDM half-row DMAs immediately (in-order completion).
  if (tid < 32) {
    const float*  base = x + (size_t)row * VQ_COLS;
    const uint32_t l0  = (uint32_t)(uintptr_t)(&s_x[0]);
    tdm_load_f32(base,           l0,               VQ_HALF);   // stage A
    tdm_load_f32(base + VQ_HALF, l0 + VQ_HALF * 4, VQ_HALF);   // stage B
  }

  // Stage this row's 16 codebook values into LDS (one cacheline).
  if (tid < VQ_NV) s_v[tid] = values[(size_t)row * VQ_NV + tid];

  // Stage A ready when TENSORcnt <= 1 (stage B may still be in flight).
  if (tid < 32) __builtin_amdgcn_s_wait_tensorcnt(1);
  __syncthreads();

  // Codebook -> registers; precompute score coefficients.
  float vv[VQ_NV], aa[VQ_NV], bb[VQ_NV];
#pragma unroll
  for (int i = 0; i < VQ_NV; ++i) {
    float v = s_v[i];
    vv[i] = v;
    aa[i] = -2.0f * v;
    bb[i] = v * v;
  }

  const float4* s_x4 = (const float4*)s_x;
  float4* out4 = (float4*)(out + (size_t)row * VQ_COLS);

  // ---- Stage A: cols [0, 1024) while TDM streams stage B into LDS ----
  {
    const int idx = tid;                     // 256 float4 = 1024 floats
    float4 xv = s_x4[idx];                   // ds_load_b128
    float4 r;
    r.x = nearest16(xv.x, vv, aa, bb);
    r.y = nearest16(xv.y, vv, aa, bb);
    r.z = nearest16(xv.z, vv, aa, bb);
    r.w = nearest16(xv.w, vv, aa, bb);
    out4[idx] = r;                           // global_store_b128
  }

  // ---- Stage B: cols [1024, 2048) ----
  if (tid < 32) __builtin_amdgcn_s_wait_tensorcnt(0);
  __syncthreads();
  {
    const int idx = tid + VQ_BLOCK;
    float4 xv = s_x4[idx];
    float4 r;
    r.x = nearest16(xv.x, vv, aa, bb);
    r.y = nearest16(xv.y, vv, aa, bb);
    r.z = nearest16(xv.z, vv, aa, bb);
    r.w = nearest16(xv.w, vv, aa, bb);
    out4[idx] = r;
  }
}

extern "C" void kernel_launch(void* const* d_in, const int* in_sizes, int n_in,
                              void* d_out, int out_size, void* d_ws, size_t ws_size,
                              hipStream_t stream) {
  const float* x      = (const float*)d_in[0];
  const float* values = (const float*)d_in[1];
  float* out          = (float*)d_out;

  const int rows = in_sizes[1] / VQ_NV;  // 4096
  vq_rows_kernel<<<rows, VQ_BLOCK, 0, stream>>>(x, values, out);
}